// GCN_21165598834728
// MI455X (gfx1250) — compile-verified
//
#include <hip/hip_runtime.h>
#include <hip/hip_bf16.h>

typedef __attribute__((ext_vector_type(2))) float v2f;
typedef __attribute__((ext_vector_type(8))) float v8f;

// ---------------------------------------------------------------------------
// Degree / normalization kernels
// ---------------------------------------------------------------------------
__global__ void gcn_deg_init(float* __restrict__ deg, int n) {
    int i = blockIdx.x * blockDim.x + threadIdx.x;
    if (i < n) deg[i] = 1.0f;   // self-loop contributes 1 to every in-degree
}

__global__ void gcn_deg_count(const long long* __restrict__ dst,
                              float* __restrict__ deg, int e) {
    int i = blockIdx.x * blockDim.x + threadIdx.x;
    if (i < e) atomicAdd(&deg[(int)dst[i]], 1.0f);
}

__global__ void gcn_rsqrt(const float* __restrict__ deg,
                          float* __restrict__ dis, int n) {
    int i = blockIdx.x * blockDim.x + threadIdx.x;
    if (i < n) {
        float d = deg[i];
        dis[i] = d > 0.0f ? rsqrtf(d) : 0.0f;
    }
}

__global__ void gcn_zero4(float* __restrict__ p, int n4) {
    int i = blockIdx.x * blockDim.x + threadIdx.x;
    if (i < n4) ((float4*)p)[i] = make_float4(0.f, 0.f, 0.f, 0.f);
}

// ---------------------------------------------------------------------------
// fp32 WMMA GEMM: C[M,N] = A[M,K] * W[K,N], row-major, K=128, N = NT*16.
// One wave per 16-row strip; 8 waves/block share W staged in LDS.
//
// W is stored in LDS PRE-PACKED in B-fragment order so each lane's B operand
// is one aligned float2 (ds_load_b64) -> feeds WMMA with no register repack:
//   packed[(g*NT + t)*32 + lane] = { W[4g+2*half + 0][t*16 + row],
//                                    W[4g+2*half + 1][t*16 + row] }
//   with half = lane>>4, row = lane&15, g = kk/4.
// Per k-group, all NT B fragments are staged into registers first (one DS
// clause + one wait), then the NT WMMAs issue back-to-back.
//
// A-frag (16x4 f32): lane<16 -> M=lane, K={kk,kk+1}; lane>=16 -> K={kk+2,kk+3}
// C/D (16x16 f32):   VGPR j : M=j (lanes 0-15), M=j+8 (lanes 16-31), N=lane&15
// ---------------------------------------------------------------------------
template <int NT>
__global__ void gcn_gemm_wmma(const float* __restrict__ A,
                              const float* __restrict__ W,
                              float* __restrict__ C, int M, int K) {
    constexpr int N      = NT * 16;
    constexpr int GROUPS = 32;                 // K/4 with K=128
    __shared__ float2 sW[GROUPS * NT * 32];    // K/2 * N float2 = K*N floats

    const int tid = threadIdx.x;
    // cooperative pack of W into fragment order
    for (int p = tid; p < GROUPS * NT * 32; p += blockDim.x) {
        int g = p / (NT * 32);
        int r = p - g * (NT * 32);
        int t = r >> 5;
        int l = r & 31;
        int k0  = 4 * g + 2 * (l >> 4);
        int col = t * 16 + (l & 15);
        sW[p] = make_float2(W[(size_t)k0 * N + col], W[(size_t)(k0 + 1) * N + col]);
    }
    __syncthreads();

    const int wave  = tid >> 5;
    const int lane  = tid & 31;
    const int strip = blockIdx.x * (blockDim.x >> 5) + wave;
    if (strip * 16 >= M) return;            // wave-uniform exit (EXEC stays full)

    const int row  = lane & 15;
    const int half = lane >> 4;             // 0: K pair {0,1}, 1: K pair {2,3}

    const float* Arow = A + (size_t)(strip * 16 + row) * K;

    v8f acc[NT];
#pragma unroll
    for (int t = 0; t < NT; ++t) acc[t] = (v8f){0.f,0.f,0.f,0.f,0.f,0.f,0.f,0.f};

#pragma unroll 4
    for (int g = 0; g < GROUPS; ++g) {
        float2 av = *(const float2*)(Arow + 4 * g + 2 * half);
        v2f a; a.x = av.x; a.y = av.y;
        const float2* bbase = sW + g * (NT * 32) + lane;

        float2 bv[NT];                       // stage all B fragments: one DS
#pragma unroll                               // clause, one dscnt wait
        for (int t = 0; t < NT; ++t) bv[t] = bbase[t * 32];

#pragma unroll
        for (int t = 0; t < NT; ++t) {
            v2f b; b.x = bv[t].x; b.y = bv[t].y;
            acc[t] = __builtin_amdgcn_wmma_f32_16x16x4_f32(
                false, a, false, b, (short)0, acc[t], false, false);
        }
    }

#pragma unroll
    for (int t = 0; t < NT; ++t) {
        float* crow = C + (size_t)strip * 16 * N + t * 16 + row;
#pragma unroll
        for (int j = 0; j < 8; ++j)
            crow[(size_t)(j + 8 * half) * N] = acc[t][j];
    }
}

// ---------------------------------------------------------------------------
// Edge aggregation: agg[dst] += h[src] * dis[src]*dis[dst].
// One wave per edge, VPT consecutive floats per lane (F = 32*VPT).
// h and agg are L2-resident (25.6 MB each << 192 MB L2).
// ---------------------------------------------------------------------------
template <int VPT>
__global__ void gcn_aggregate(const float* __restrict__ h,
                              const long long* __restrict__ src,
                              const long long* __restrict__ dst,
                              const float* __restrict__ dis,
                              float* __restrict__ agg, int E) {
    constexpr int F = 32 * VPT;
    int e = blockIdx.x * (blockDim.x >> 5) + (threadIdx.x >> 5);
    if (e >= E) return;
    int lane = threadIdx.x & 31;

    int s = (int)src[e];
    int d = (int)dst[e];
    float nrm = dis[s] * dis[d];

    const float* hp = h   + (size_t)s * F + lane * VPT;
    float*       ap = agg + (size_t)d * F + lane * VPT;

    if constexpr (VPT == 4) {
        float4 v = *(const float4*)hp;
        atomicAdd(ap + 0, v.x * nrm);
        atomicAdd(ap + 1, v.y * nrm);
        atomicAdd(ap + 2, v.z * nrm);
        atomicAdd(ap + 3, v.w * nrm);
    } else {
        float2 v = *(const float2*)hp;
        atomicAdd(ap + 0, v.x * nrm);
        atomicAdd(ap + 1, v.y * nrm);
    }
}

// ---------------------------------------------------------------------------
// Self-loop + bias + ReLU (in place on agg)
// ---------------------------------------------------------------------------
template <int F>
__global__ void gcn_finalize_relu(const float* __restrict__ h,
                                  const float* __restrict__ b,
                                  const float* __restrict__ dis,
                                  float* __restrict__ agg, int n) {
    int i = blockIdx.x * blockDim.x + threadIdx.x;
    if (i >= n * F) return;
    int node = i / F;
    int f    = i % F;
    float ds = dis[node];
    float v  = agg[i] + h[i] * ds * ds + b[f];
    agg[i]   = fmaxf(v, 0.0f);
}

// ---------------------------------------------------------------------------
// Final layer: self-loop + bias + log_softmax over 64 classes. Wave per node,
// 2 logits per lane, wave32 shuffle reductions.
// ---------------------------------------------------------------------------
__global__ void gcn_logsoftmax64(const float* __restrict__ h,
                                 const float* __restrict__ agg,
                                 const float* __restrict__ dis,
                                 const float* __restrict__ b,
                                 float* __restrict__ out, int n) {
    int node = blockIdx.x * (blockDim.x >> 5) + (threadIdx.x >> 5);
    if (node >= n) return;
    int lane = threadIdx.x & 31;

    float ds = dis[node];
    size_t base = (size_t)node * 64;
    float v0 = agg[base + lane]      + h[base + lane]      * ds * ds + b[lane];
    float v1 = agg[base + 32 + lane] + h[base + 32 + lane] * ds * ds + b[32 + lane];

    float m = fmaxf(v0, v1);
#pragma unroll
    for (int o = 16; o > 0; o >>= 1) m = fmaxf(m, __shfl_xor(m, o, 32));
    float s = __expf(v0 - m) + __expf(v1 - m);
#pragma unroll
    for (int o = 16; o > 0; o >>= 1) s += __shfl_xor(s, o, 32);
    float lse = m + __logf(s);

    out[base + lane]      = v0 - lse;
    out[base + 32 + lane] = v1 - lse;
}

// ---------------------------------------------------------------------------
// Orchestration
// ---------------------------------------------------------------------------
extern "C" void kernel_launch(void* const* d_in, const int* in_sizes, int n_in,
                              void* d_out, int out_size, void* d_ws, size_t ws_size,
                              hipStream_t stream) {
    const float*      x   = (const float*)d_in[0];   // [N,128]
    const float*      W1  = (const float*)d_in[1];   // [128,128]
    const float*      b1  = (const float*)d_in[2];   // [128]
    const float*      W2  = (const float*)d_in[3];   // [128,128]
    const float*      b2  = (const float*)d_in[4];   // [128]
    const float*      W3  = (const float*)d_in[5];   // [128,64]
    const float*      b3  = (const float*)d_in[6];   // [64]
    const long long*  ei  = (const long long*)d_in[7]; // [2,E]

    const int N = in_sizes[0] / 128;   // 50000
    const int E = in_sizes[7] / 2;     // 800000
    const long long* src = ei;
    const long long* dst = ei + E;

    float* tmpA = (float*)d_ws;                 // [N,128]  h buffer
    float* tmpB = tmpA + (size_t)N * 128;       // [N,128]  agg / next input
    float* tmpC = tmpB + (size_t)N * 128;       // [N,64]   layer-3 h
    float* deg  = tmpC + (size_t)N * 64;        // [N]
    float* dis  = deg + N;                      // [N]

    const int T = 256;
    // --- degrees & symmetric norm --------------------------------------
    gcn_deg_init <<<(N + T - 1) / T, T, 0, stream>>>(deg, N);
    gcn_deg_count<<<(E + T - 1) / T, T, 0, stream>>>(dst, deg, E);
    gcn_rsqrt    <<<(N + T - 1) / T, T, 0, stream>>>(deg, dis, N);

    const int strips     = N / 16;                     // 3125
    const int gemmBlocks = (strips + 7) / 8;           // 8 waves / block
    const int aggBlocks  = (E + 7) / 8;                // wave per edge
    const int z128       = (N * 128 / 4 + T - 1) / T;
    const int z64        = (N * 64 / 4 + T - 1) / T;
    const int f128       = (N * 128 + T - 1) / T;

    // --- layer 1 -------------------------------------------------------
    gcn_gemm_wmma<8><<<gemmBlocks, T, 0, stream>>>(x, W1, tmpA, N, 128);
    gcn_zero4       <<<z128, T, 0, stream>>>(tmpB, N * 128 / 4);
    gcn_aggregate<4><<<aggBlocks, T, 0, stream>>>(tmpA, src, dst, dis, tmpB, E);
    gcn_finalize_relu<128><<<f128, T, 0, stream>>>(tmpA, b1, dis, tmpB, N);

    // --- layer 2 -------------------------------------------------------
    gcn_gemm_wmma<8><<<gemmBlocks, T, 0, stream>>>(tmpB, W2, tmpA, N, 128);
    gcn_zero4       <<<z128, T, 0, stream>>>(tmpB, N * 128 / 4);
    gcn_aggregate<4><<<aggBlocks, T, 0, stream>>>(tmpA, src, dst, dis, tmpB, E);
    gcn_finalize_relu<128><<<f128, T, 0, stream>>>(tmpA, b2, dis, tmpB, N);

    // --- layer 3 + log_softmax ----------------------------------------
    gcn_gemm_wmma<4><<<gemmBlocks, T, 0, stream>>>(tmpB, W3, tmpC, N, 128);
    gcn_zero4       <<<z64, T, 0, stream>>>(tmpA, N * 64 / 4);
    gcn_aggregate<2><<<aggBlocks, T, 0, stream>>>(tmpC, src, dst, dis, tmpA, E);
    gcn_logsoftmax64<<<(N + 7) / 8, T, 0, stream>>>(tmpC, tmpA, dis, b3,
                                                    (float*)d_out, N);
}